// PyFlowODE_2705829396896
// MI455X (gfx1250) — compile-verified
//
#include <hip/hip_runtime.h>
#include <math.h>

typedef __attribute__((ext_vector_type(2))) float v2f;
typedef __attribute__((ext_vector_type(8))) float v8f;

#define NQ      2048
#define NB      4096
#define DIM     8
#define TSTEPS  16
#define QTILE   16
#define WAVES   8
#define THREADS 256
#define TILES_PER_WAVE ((NB / WAVES) / 16)   // 32 sixteen-bank tiles per wave

// lane i <-> lane i^16 exchange (group-of-32 swizzle, xor=0x10, and=0x1f)
__device__ __forceinline__ float swapx16(float x) {
  return __int_as_float(__builtin_amdgcn_ds_swizzle(__float_as_int(x), 0x401f));
}

__global__ __launch_bounds__(THREADS, 1)
void flow_ode_kernel(const float* __restrict__ x1,
                     const float* __restrict__ x0,
                     float* __restrict__ out) {
  __shared__ float lx0[NB * DIM];          // 128 KB bank, resident all 16 steps
  __shared__ float ln0[NB];                // 16 KB  ||x0||^2
  __shared__ float lmrg[WAVES][QTILE][10]; // per-step flash-softmax partials

  const int  tid  = threadIdx.x;
  const int  w    = tid >> 5;
  const int  lane = tid & 31;
  const int  qn   = lane & 15;
  const bool lo   = lane < 16;
  const int  hb   = lo ? 0 : 8;            // bank-half offset for hi lanes

  // ---- stage x0 into LDS, compute ||x0||^2 ----
  {
    const float4* g = (const float4*)x0;
    float4* l = (float4*)lx0;
    for (int i = tid; i < NB * DIM / 4; i += THREADS) l[i] = g[i];
  }
  __syncthreads();
  for (int b = tid; b < NB; b += THREADS) {
    float s = 0.f;
    #pragma unroll
    for (int d = 0; d < DIM; ++d) { float v = lx0[b * DIM + d]; s += v * v; }
    ln0[b] = s;
  }
  __syncthreads();

  const float h = 1.0f / (float)TSTEPS;
  const int qbase = blockIdx.x * QTILE;

  // every lane redundantly carries query (qn): lane n and n+16 hold row n
  float xt[DIM], Gp[DIM];
  {
    const float4* g = (const float4*)(x1 + (size_t)(qbase + qn) * DIM);
    float4 a = g[0], b = g[1];
    xt[0] = a.x; xt[1] = a.y; xt[2] = a.z; xt[3] = a.w;
    xt[4] = b.x; xt[5] = b.y; xt[6] = b.z; xt[7] = b.w;
  }
  #pragma unroll
  for (int d = 0; d < DIM; ++d) Gp[d] = 0.f;

  for (int e = 0; e < TSTEPS; ++e) {
    const float t     = 1.0f - (float)e * h;   // 1, 15/16, ..., 1/16
    const float alpha = 1.0f - t;
    const float it2   = 1.0f / (t * t);
    const float sB    = alpha * it2;                    // scale on Q side
    const float sC    = -0.5f * alpha * alpha * it2;    // bias scale on ||x0||^2

    // Heun predictor point y = xt - G_prev*h  (== x1 at e=0 since G_prev=0)
    float y[DIM];
    #pragma unroll
    for (int d = 0; d < DIM; ++d) y[d] = xt[d] - Gp[d] * h;

    // B operand of S^T = x0 * y^T : B[k][n] = sB*y[n][k]
    // vgpr0 = rows k,k+2 (lo/hi lane halves), vgpr1 = rows k+1,k+3
    v2f B0, B1;
    B0.x = sB * (lo ? y[0] : y[2]);
    B0.y = sB * (lo ? y[1] : y[3]);
    B1.x = sB * (lo ? y[4] : y[6]);
    B1.y = sB * (lo ? y[5] : y[7]);

    // ---- online softmax over this wave's 512 banks ----
    float m = -INFINITY, ss = 0.f;
    float acc[DIM];
    #pragma unroll
    for (int d = 0; d < DIM; ++d) acc[d] = 0.f;

    for (int tt = 0; tt < TILES_PER_WAVE; ++tt) {
      const int tb = (w * TILES_PER_WAVE + tt) * 16;

      // A = x0 tile (16 banks x 8), f32 A-layout: lane=bank row, hi lanes k+2
      const float* ar = &lx0[(size_t)(tb + qn) * DIM + (lo ? 0 : 2)];
      v2f A0, A1;
      A0.x = ar[0]; A0.y = ar[1];   // k = 0,1 (lo) / 2,3 (hi)
      A1.x = ar[4]; A1.y = ar[5];   // k = 4,5 (lo) / 6,7 (hi)

      // bias folded into accumulator: C[bank][q] = sC*||x0[bank]||^2
      v8f c;
      #pragma unroll
      for (int v = 0; v < 8; ++v) c[v] = sC * ln0[tb + hb + v];

      v8f dm = __builtin_amdgcn_wmma_f32_16x16x4_f32(
          false, A0, false, B0, (short)0, c,  false, false);
      dm     = __builtin_amdgcn_wmma_f32_16x16x4_f32(
          false, A1, false, B1, (short)0, dm, false, false);

      // D layout: vgpr v, lane n => logit[bank tb+v+hb][query n]
      float tmax = dm[0];
      #pragma unroll
      for (int v = 1; v < 8; ++v) tmax = fmaxf(tmax, dm[v]);
      const float mn = fmaxf(m, tmax);
      const float r  = __expf(m - mn);        // 0 on first tile (m = -inf)
      ss *= r;
      #pragma unroll
      for (int d = 0; d < DIM; ++d) acc[d] *= r;
      m = mn;

      float p[8];
      #pragma unroll
      for (int v = 0; v < 8; ++v) { p[v] = __expf(dm[v] - m); ss += p[v]; }

      #pragma unroll
      for (int v = 0; v < 8; ++v) {
        const float4* xr = (const float4*)&lx0[(size_t)(tb + hb + v) * DIM];
        const float4 xa = xr[0], xb = xr[1];
        acc[0] += p[v] * xa.x; acc[1] += p[v] * xa.y;
        acc[2] += p[v] * xa.z; acc[3] += p[v] * xa.w;
        acc[4] += p[v] * xb.x; acc[5] += p[v] * xb.y;
        acc[6] += p[v] * xb.z; acc[7] += p[v] * xb.w;
      }
    }

    // ---- merge lane halves (banks 0-7 vs 8-15 of each tile) ----
    {
      const float m2 = swapx16(m), s2 = swapx16(ss);
      const float mm = fmaxf(m, m2);
      const float f1 = __expf(m - mm), f2 = __expf(m2 - mm);
      const float S  = ss * f1 + s2 * f2;
      float A[DIM];
      #pragma unroll
      for (int d = 0; d < DIM; ++d) {
        const float o = swapx16(acc[d]);
        A[d] = acc[d] * f1 + o * f2;
      }
      m = mm; ss = S;
      #pragma unroll
      for (int d = 0; d < DIM; ++d) acc[d] = A[d];
    }

    if (lo) {
      float* mp = &lmrg[w][qn][0];
      mp[0] = m; mp[1] = ss;
      #pragma unroll
      for (int d = 0; d < DIM; ++d) mp[2 + d] = acc[d];
    }
    __syncthreads();

    // ---- cross-wave merge, done redundantly by every lane (keeps state
    //      replicated so next step's B build is register-only) ----
    float M = -INFINITY, SS = 0.f, Q[DIM];
    #pragma unroll
    for (int d = 0; d < DIM; ++d) Q[d] = 0.f;
    for (int ww = 0; ww < WAVES; ++ww) {
      const float* mp = &lmrg[ww][qn][0];
      const float mo = mp[0], so = mp[1];
      const float mx = fmaxf(M, mo);
      const float fa = __expf(M - mx), fb = __expf(mo - mx);
      SS = SS * fa + so * fb;
      #pragma unroll
      for (int d = 0; d < DIM; ++d) Q[d] = Q[d] * fa + mp[2 + d] * fb;
      M = mx;
    }

    const float invS = 1.0f / SS;
    const float upd  = (e == 0) ? 0.f : (0.5f * h);  // e=0 only seeds G_prev
    #pragma unroll
    for (int d = 0; d < DIM; ++d) {
      const float q  = Q[d] * invS;
      const float Gn = (y[d] - q) / t;
      xt[d] -= (Gp[d] + Gn) * upd;
      Gp[d]  = Gn;
    }
    __syncthreads();   // protect lmrg before next step overwrites it
  }

  if (tid < 16) {
    float4* ov = (float4*)(out + (size_t)(qbase + qn) * DIM);
    ov[0] = make_float4(xt[0], xt[1], xt[2], xt[3]);
    ov[1] = make_float4(xt[4], xt[5], xt[6], xt[7]);
  }
}

extern "C" void kernel_launch(void* const* d_in, const int* in_sizes, int n_in,
                              void* d_out, int out_size, void* d_ws, size_t ws_size,
                              hipStream_t stream) {
  const float* x1 = (const float*)d_in[0];   // (2048, 8)
  const float* x0 = (const float*)d_in[1];   // (4096, 8)
  float* out = (float*)d_out;                // (2048, 8)
  const int nq = in_sizes[0] / DIM;
  const int blocks = nq / QTILE;             // 128
  flow_ode_kernel<<<blocks, THREADS, 0, stream>>>(x1, x0, out);
}